// GradientPurifier_32220844654770
// MI455X (gfx1250) — compile-verified
//
#include <hip/hip_runtime.h>

#define NUM_CLASSES 1000
#define DIM 256
#define MOM 0.9f

#define SLICE_W 16                    // columns per segment-sum slice
#define NSLICES (DIM / SLICE_W)       // 16
#define ROW_CHUNKS 32

typedef __attribute__((ext_vector_type(2))) float v2f;
typedef __attribute__((ext_vector_type(8))) float v8f;

// ---------------------------------------------------------------------------
// 0) zero the fp32 accumulators in workspace (sums + counts)
// ---------------------------------------------------------------------------
__global__ void zero_kernel(float* __restrict__ p, int n) {
    int i = blockIdx.x * blockDim.x + threadIdx.x;
    if (i < n) p[i] = 0.0f;
}

// ---------------------------------------------------------------------------
// 1) segment sum: grid (NSLICES, ROW_CHUNKS). Each block owns a 16-column
//    slice and a row chunk, privatizes [1000 x 16] fp32 in LDS (64 KB of the
//    WGP's 320 KB), ds_add_f32 per element, then flushes via global fp32
//    atomics. Wave handles 2 rows x 16 cols -> 2x64B coalesced reads.
// ---------------------------------------------------------------------------
__global__ __launch_bounds__(256) void seg_sum_kernel(
    const float* __restrict__ g, const int* __restrict__ lab,
    float* __restrict__ sums, int n, int rows_per_chunk) {
    __shared__ float acc[NUM_CLASSES * SLICE_W];   // 64000 B
    const int slice = blockIdx.x;                  // 0..15
    for (int i = threadIdx.x; i < NUM_CLASSES * SLICE_W; i += blockDim.x)
        acc[i] = 0.0f;
    __syncthreads();

    const int lane = threadIdx.x & 31;
    const int wave = threadIdx.x >> 5;             // 0..7
    const int sub  = lane >> 4;                    // 0/1 : row within pair
    const int col  = lane & 15;                    // column within slice

    const int row0    = blockIdx.y * rows_per_chunk;
    const int row_end = min(row0 + rows_per_chunk, n);
    const int colbase = slice * SLICE_W + col;

    for (int r = row0 + wave * 2 + sub; r < row_end; r += 16) {
        int cls   = lab[r];
        float val = g[(size_t)r * DIM + colbase];
        atomicAdd(&acc[cls * SLICE_W + col], val);     // ds_add_f32
    }
    __syncthreads();

    for (int i = threadIdx.x; i < NUM_CLASSES * SLICE_W; i += blockDim.x) {
        float s = acc[i];
        if (s != 0.0f) {
            int cls = i / SLICE_W;
            int cc  = i - cls * SLICE_W;
            atomicAdd(&sums[(size_t)cls * DIM + slice * SLICE_W + cc], s);
        }
    }
}

// ---------------------------------------------------------------------------
// 2) label histogram with LDS privatization
// ---------------------------------------------------------------------------
__global__ __launch_bounds__(256) void count_kernel(
    const int* __restrict__ lab, float* __restrict__ counts, int n) {
    __shared__ float c[NUM_CLASSES];
    for (int i = threadIdx.x; i < NUM_CLASSES; i += blockDim.x) c[i] = 0.0f;
    __syncthreads();
    for (int r = blockIdx.x * blockDim.x + threadIdx.x; r < n;
         r += gridDim.x * blockDim.x)
        atomicAdd(&c[lab[r]], 1.0f);
    __syncthreads();
    for (int i = threadIdx.x; i < NUM_CLASSES; i += blockDim.x)
        if (c[i] != 0.0f) atomicAdd(&counts[i], c[i]);
}

// ---------------------------------------------------------------------------
// 3) EMA centroid update
// ---------------------------------------------------------------------------
__global__ __launch_bounds__(256) void ema_kernel(
    const float* __restrict__ sums, const float* __restrict__ counts,
    const float* __restrict__ cent, float* __restrict__ nc) {
    int c = blockIdx.x, d = threadIdx.x;
    float cnt  = counts[c];
    float s    = sums[(size_t)c * DIM + d];
    float ce   = cent[(size_t)c * DIM + d];
    float mean = s / fmaxf(cnt, 1.0f);
    nc[(size_t)c * DIM + d] = (cnt > 0.0f) ? (MOM * ce + (1.0f - MOM) * mean) : ce;
}

// ---------------------------------------------------------------------------
// 4) column mean over classes (one block, coalesced column reads)
// ---------------------------------------------------------------------------
__global__ __launch_bounds__(256) void colmean_kernel(
    const float* __restrict__ nc, float* __restrict__ m) {
    int d = threadIdx.x;
    float s = 0.0f;
    for (int c = 0; c < NUM_CLASSES; ++c) s += nc[(size_t)c * DIM + d];
    m[d] = s * (1.0f / NUM_CLASSES);
}

// ---------------------------------------------------------------------------
// 5) Gram matrix G = centered^T * centered via V_WMMA_F32_16X16X4_F32.
//    One wave (32 threads) per 16x16 tile of G; K = 1000 in steps of 4.
//    Fragment layouts per ISA 7.12.2 (32-bit A 16x4, 32-bit B 4x16, C/D 8 VGPR).
//    Mean subtraction fused into the fragment loads.
// ---------------------------------------------------------------------------
__global__ __launch_bounds__(32) void gram_wmma_kernel(
    const float* __restrict__ nc, const float* __restrict__ m,
    float* __restrict__ G) {
    const int ti   = blockIdx.x;       // tile row of G (d1)
    const int tj   = blockIdx.y;       // tile col of G (d2)
    const int lane = threadIdx.x;      // 0..31, full wave -> EXEC all 1s
    const int half = lane >> 4;        // 0: K pair {0,1}; 1: K pair {2,3}
    const int lidx = lane & 15;

    const int cola = ti * 16 + lidx;
    const int colb = tj * 16 + lidx;
    const float ma = m[cola];
    const float mb = m[colb];

    v8f acc = {};
    for (int k0 = 0; k0 < NUM_CLASSES; k0 += 4) {
        const int ka = k0 + 2 * half;
        v2f a, b;
        // A (16x4): lanes 0-15 hold K=0 (v0), K=1 (v1); lanes 16-31 hold K=2,3
        a.x = nc[(size_t)(ka + 0) * DIM + cola] - ma;
        a.y = nc[(size_t)(ka + 1) * DIM + cola] - ma;
        // B (4x16): row-striped across lanes, same half split
        b.x = nc[(size_t)(ka + 0) * DIM + colb] - mb;
        b.y = nc[(size_t)(ka + 1) * DIM + colb] - mb;
        acc = __builtin_amdgcn_wmma_f32_16x16x4_f32(
            /*neg_a=*/false, a, /*neg_b=*/false, b,
            /*c_mod=*/(short)0, acc, /*reuse_a=*/false, /*reuse_b=*/false);
    }
#pragma unroll
    for (int r = 0; r < 8; ++r) {
        // C/D layout: VGPR r holds M = r (lanes 0-15) / M = r+8 (lanes 16-31)
        G[(size_t)(ti * 16 + r + 8 * half) * DIM + tj * 16 + lidx] = acc[r];
    }
}

// ---------------------------------------------------------------------------
// 6) power iteration on G (256x256, PSD) -> top eigenvector == top right
//    singular vector of centered. Sign is irrelevant (v v^T projection).
// ---------------------------------------------------------------------------
__global__ __launch_bounds__(256) void power_iter_kernel(
    const float* __restrict__ G, float* __restrict__ vout) {
    __shared__ float vs[DIM];
    __shared__ float red[DIM];
    const int d = threadIdx.x;
    vs[d] = 1.0f;
    __syncthreads();
    for (int it = 0; it < 64; ++it) {
        const float* row = G + (size_t)d * DIM;
        float dot = 0.0f;
#pragma unroll 8
        for (int e = 0; e < DIM; ++e) dot += row[e] * vs[e];
        red[d] = dot * dot;
        __syncthreads();
        for (int s = DIM / 2; s > 0; s >>= 1) {
            if (d < s) red[d] += red[d + s];
            __syncthreads();
        }
        float nrm = rsqrtf(fmaxf(red[0], 1e-30f));
        __syncthreads();
        vs[d] = dot * nrm;
        __syncthreads();
    }
    vout[d] = vs[d];
}

// ---------------------------------------------------------------------------
// 7) projection pass: out = g - (g . v) v. One wave per row; two coalesced
//    float4 loads per lane (512B segments per wave), wave shfl_xor reduction.
//    This is the bandwidth-critical kernel (820 MB of the ~1.25 GB total).
// ---------------------------------------------------------------------------
__global__ __launch_bounds__(256) void purify_kernel(
    const float* __restrict__ g, const float* __restrict__ v,
    float* __restrict__ out, int n) {
    __shared__ float vs[DIM];
    if (threadIdx.x < DIM) vs[threadIdx.x] = v[threadIdx.x];
    __syncthreads();

    const int lane = threadIdx.x & 31;
    const int wave = threadIdx.x >> 5;
    const int row  = blockIdx.x * 8 + wave;
    if (row >= n) return;

    const float4* grow = (const float4*)(g + (size_t)row * DIM);
    float4*       orow = (float4*)(out + (size_t)row * DIM);

    float4 a0 = grow[lane];          // floats [4*lane   .. 4*lane+3]
    float4 a1 = grow[lane + 32];     // floats [128+4*lane .. ]
    const int o0 = lane * 4;
    const int o1 = 128 + lane * 4;

    float dot = a0.x * vs[o0]     + a0.y * vs[o0 + 1]
              + a0.z * vs[o0 + 2] + a0.w * vs[o0 + 3]
              + a1.x * vs[o1]     + a1.y * vs[o1 + 1]
              + a1.z * vs[o1 + 2] + a1.w * vs[o1 + 3];
#pragma unroll
    for (int off = 16; off > 0; off >>= 1)
        dot += __shfl_xor(dot, off, 32);

    float4 r0, r1;
    r0.x = a0.x - dot * vs[o0];     r0.y = a0.y - dot * vs[o0 + 1];
    r0.z = a0.z - dot * vs[o0 + 2]; r0.w = a0.w - dot * vs[o0 + 3];
    r1.x = a1.x - dot * vs[o1];     r1.y = a1.y - dot * vs[o1 + 1];
    r1.z = a1.z - dot * vs[o1 + 2]; r1.w = a1.w - dot * vs[o1 + 3];
    orow[lane]      = r0;
    orow[lane + 32] = r1;
}

// ---------------------------------------------------------------------------
// launcher
// ---------------------------------------------------------------------------
extern "C" void kernel_launch(void* const* d_in, const int* in_sizes, int n_in,
                              void* d_out, int out_size, void* d_ws, size_t ws_size,
                              hipStream_t stream) {
    const float* grads  = (const float*)d_in[0];   // [N, 256]
    const int*   labels = (const int*)  d_in[1];   // [N]
    const float* cents  = (const float*)d_in[2];   // [1000, 256]
    float*       out    = (float*)d_out;           // [N, 256]
    const int N = in_sizes[0] / DIM;

    // workspace layout (floats)
    float* ws      = (float*)d_ws;
    float* sums    = ws;                                   // 1000*256
    float* counts  = sums   + NUM_CLASSES * DIM;           // 1000
    float* nc      = counts + NUM_CLASSES;                 // 1000*256
    float* colmean = nc     + NUM_CLASSES * DIM;           // 256
    float* G       = colmean + DIM;                        // 256*256
    float* vvec    = G      + DIM * DIM;                   // 256

    // 0) zero accumulators (sums + counts)
    {
        int nz = NUM_CLASSES * DIM + NUM_CLASSES;
        zero_kernel<<<(nz + 255) / 256, 256, 0, stream>>>(sums, nz);
    }
    // 1) segment sum
    {
        int rows_per_chunk = (N + ROW_CHUNKS - 1) / ROW_CHUNKS;
        dim3 grid(NSLICES, ROW_CHUNKS);
        seg_sum_kernel<<<grid, 256, 0, stream>>>(grads, labels, sums, N,
                                                 rows_per_chunk);
    }
    // 2) counts
    count_kernel<<<256, 256, 0, stream>>>(labels, counts, N);
    // 3) EMA update
    ema_kernel<<<NUM_CLASSES, DIM, 0, stream>>>(sums, counts, cents, nc);
    // 4) column mean
    colmean_kernel<<<1, DIM, 0, stream>>>(nc, colmean);
    // 5) Gram matrix via WMMA (16x16 tiles, one wave each)
    {
        dim3 grid(DIM / 16, DIM / 16);
        gram_wmma_kernel<<<grid, 32, 0, stream>>>(nc, colmean, G);
    }
    // 6) top eigenvector by power iteration
    power_iter_kernel<<<1, DIM, 0, stream>>>(G, vvec);
    // 7) rank-1 projection removal
    purify_kernel<<<(N + 7) / 8, 256, 0, stream>>>(grads, vvec, out, N);
}